// RangePredictor_63333587747658
// MI455X (gfx1250) — compile-verified
//
#include <hip/hip_runtime.h>
#include <hip/hip_bf16.h>
#include <math.h>

typedef float v2f __attribute__((ext_vector_type(2)));
typedef float v8f __attribute__((ext_vector_type(8)));

// ---------------------------------------------------------------------------
// Generic fill
// ---------------------------------------------------------------------------
__global__ void fill_kernel(float* __restrict__ p, float v, long long n) {
    long long i = (long long)blockIdx.x * blockDim.x + threadIdx.x;
    if (i < n) p[i] = v;
}

// ---------------------------------------------------------------------------
// Degree accumulation: deg[dst[e]] += 1
// ---------------------------------------------------------------------------
__global__ void degree_kernel(const int* __restrict__ dst, float* __restrict__ deg,
                              long long E) {
    long long e = (long long)blockIdx.x * blockDim.x + threadIdx.x;
    if (e < E) unsafeAtomicAdd(&deg[dst[e]], 1.0f);
}

// inv[i] = rsqrt(deg[i] + 1)   (+1 self loop)
__global__ void rsqrt_kernel(float* __restrict__ inv, int N) {
    int i = blockIdx.x * blockDim.x + threadIdx.x;
    if (i < N) inv[i] = rsqrtf(inv[i] + 1.0f);
}

// ---------------------------------------------------------------------------
// f32 WMMA GEMM: C[N,OUT] = A[N,K] @ W[K,OUT]
// One wave computes a 32-row x 64-col strip (2 row-tiles x 4 col-tiles).
// Main K-loop (K & ~3) is branch-free with pointer-increment addressing;
// the K%4 tail uses clamped-index + mask-multiply loads (no EXEC divergence,
// WMMA always sees EXEC = all ones).
// grid.x = ceil(N/32), grid.y = OUT/64, block = 32.
// ---------------------------------------------------------------------------
__global__ __launch_bounds__(32)
void gemm_f32_wmma(const float* __restrict__ A, const float* __restrict__ W,
                   float* __restrict__ C, int N, int K, int OUT) {
    const int lane  = threadIdx.x;           // 0..31
    const int row0  = blockIdx.x * 32;
    const int col0  = blockIdx.y * 64;
    const int mrow  = lane & 15;             // M (and N) index of this lane
    const int khalf = (lane >> 4) * 2;       // 0 for lanes 0-15, 2 for 16-31

    int r0 = row0 + mrow;       if (r0 >= N) r0 = N - 1;   // clamp (stores guarded)
    int r1 = row0 + 16 + mrow;  if (r1 >= N) r1 = N - 1;

    const float* a0p = A + (size_t)r0 * K + khalf;
    const float* a1p = A + (size_t)r1 * K + khalf;
    const float* wp  = W + (size_t)khalf * OUT + (col0 + mrow);

    v8f acc0[4], acc1[4];
#pragma unroll
    for (int t = 0; t < 4; ++t) { acc0[t] = (v8f){}; acc1[t] = (v8f){}; }

    const int kmain = K & ~3;
    for (int k = 0; k < kmain; k += 4) {
        v2f a0, a1;
        a0.x = a0p[0]; a0.y = a0p[1];
        a1.x = a1p[0]; a1.y = a1p[1];
#pragma unroll
        for (int t = 0; t < 4; ++t) {
            v2f b;
            b.x = wp[t * 16];
            b.y = wp[t * 16 + OUT];
            acc0[t] = __builtin_amdgcn_wmma_f32_16x16x4_f32(
                false, a0, false, b, (short)0, acc0[t], false, false);
            acc1[t] = __builtin_amdgcn_wmma_f32_16x16x4_f32(
                false, a1, false, b, (short)0, acc1[t], false, false);
        }
        a0p += 4; a1p += 4; wp += (size_t)4 * OUT;
    }

    if (K & 3) {  // uniform tail (layer 1: K=5)
        const int kk0 = kmain + khalf, kk1 = kk0 + 1;
        const float m0 = (kk0 < K) ? 1.0f : 0.0f;
        const float m1 = (kk1 < K) ? 1.0f : 0.0f;
        const int   i0 = (kk0 < K) ? kk0 : (K - 1);
        const int   i1 = (kk1 < K) ? kk1 : (K - 1);
        v2f a0, a1;
        a0.x = A[(size_t)r0 * K + i0] * m0;
        a0.y = A[(size_t)r0 * K + i1] * m1;
        a1.x = A[(size_t)r1 * K + i0] * m0;
        a1.y = A[(size_t)r1 * K + i1] * m1;
#pragma unroll
        for (int t = 0; t < 4; ++t) {
            const int col = col0 + t * 16 + mrow;
            v2f b;
            b.x = W[(size_t)i0 * OUT + col] * m0;
            b.y = W[(size_t)i1 * OUT + col] * m1;
            acc0[t] = __builtin_amdgcn_wmma_f32_16x16x4_f32(
                false, a0, false, b, (short)0, acc0[t], false, false);
            acc1[t] = __builtin_amdgcn_wmma_f32_16x16x4_f32(
                false, a1, false, b, (short)0, acc1[t], false, false);
        }
    }

    // D layout: VGPR v -> M = v + 8*(lane>=16), N = lane&15 (per 16x16 tile)
    const int rofs = (lane >> 4) * 8;
#pragma unroll
    for (int t = 0; t < 4; ++t) {
        const int col = col0 + t * 16 + mrow;
#pragma unroll
        for (int v = 0; v < 8; ++v) {
            int r = row0 + rofs + v;
            if (r < N) C[(size_t)r * OUT + col] = acc0[t][v];
            r += 16;
            if (r < N) C[(size_t)r * OUT + col] = acc1[t][v];
        }
    }
}

// ---------------------------------------------------------------------------
// Edge scatter: agg[dst] += inv[src]*inv[dst] * hw[src]
// 2^cshift float4-chunks per edge; one thread per (edge, chunk).
// ---------------------------------------------------------------------------
__global__ void scatter_kernel(const float4* __restrict__ hw,
                               const int* __restrict__ src,
                               const int* __restrict__ dst,
                               const float* __restrict__ inv,
                               float* __restrict__ agg,
                               long long E, int cshift) {
    long long gid = (long long)blockIdx.x * blockDim.x + threadIdx.x;
    long long total = E << cshift;
    if (gid >= total) return;
    long long e = gid >> cshift;
    int c = (int)(gid & ((1 << cshift) - 1));
    int s = src[e], d = dst[e];
    float w = inv[s] * inv[d];
    float4 m = hw[((size_t)s << cshift) + c];
    float* ap = agg + ((((size_t)d << cshift) + c) << 2);
    unsafeAtomicAdd(ap + 0, m.x * w);
    unsafeAtomicAdd(ap + 1, m.y * w);
    unsafeAtomicAdd(ap + 2, m.z * w);
    unsafeAtomicAdd(ap + 3, m.w * w);
}

// ---------------------------------------------------------------------------
// Finalize: agg = act(agg + hw*inv^2 + b)
// ---------------------------------------------------------------------------
__global__ void finalize_kernel(float* __restrict__ agg, const float* __restrict__ hw,
                                const float* __restrict__ inv, const float* __restrict__ b,
                                long long total, int wshift, int relu) {
    long long i = (long long)blockIdx.x * blockDim.x + threadIdx.x;
    if (i >= total) return;
    int node = (int)(i >> wshift);
    int j    = (int)(i & ((1 << wshift) - 1));
    float iv = inv[node];
    float v = agg[i] + hw[i] * iv * iv + b[j];
    if (relu) v = fmaxf(v, 0.0f);
    agg[i] = v;
}

// ---------------------------------------------------------------------------
// Mean-pool reduce (width 128): pool[j] += sum_n h[n,j]
// ---------------------------------------------------------------------------
__global__ void pool_reduce(const float* __restrict__ h, float* __restrict__ pool, int N) {
    int j = threadIdx.x;  // 0..127
    float s = 0.0f;
    for (int n = blockIdx.x; n < N; n += gridDim.x)
        s += h[((size_t)n << 7) + j];
    unsafeAtomicAdd(&pool[j], s);
}

// ---------------------------------------------------------------------------
// Global-feature reductions over x[N,5]
// out6: [sum x2, sum x3, sum x4, cnt(x2==1), sum(mask*x0), sum(mask*x1)]
// ---------------------------------------------------------------------------
__global__ void gfeat_reduce(const float* __restrict__ x, float* __restrict__ out6, int N) {
    int t = blockIdx.x * blockDim.x + threadIdx.x;
    int stride = gridDim.x * blockDim.x;
    float s2 = 0, s3 = 0, s4 = 0, cnt = 0, sl = 0, sm = 0;
    for (int n = t; n < N; n += stride) {
        const float* r = x + (size_t)n * 5;
        float x0 = r[0], x1 = r[1], x2 = r[2], x3 = r[3], x4 = r[4];
        s2 += x2; s3 += x3; s4 += x4;
        if (x2 == 1.0f) { cnt += 1.0f; sl += x0; sm += x1; }
    }
    unsafeAtomicAdd(&out6[0], s2);
    unsafeAtomicAdd(&out6[1], s3);
    unsafeAtomicAdd(&out6[2], s4);
    unsafeAtomicAdd(&out6[3], cnt);
    unsafeAtomicAdd(&out6[4], sl);
    unsafeAtomicAdd(&out6[5], sm);
}

// ---------------------------------------------------------------------------
// Head: emb[135] = [pool/N | gfeat], MLP 135->64->32->4, softplus + transforms.
// scratch: [0..5]=gfeat sums, [8..135]=pool sums.  Single block of 160.
// ---------------------------------------------------------------------------
__global__ void head_kernel(const float* __restrict__ scratch,
                            const float* __restrict__ Wp1, const float* __restrict__ bp1,
                            const float* __restrict__ Wp2, const float* __restrict__ bp2,
                            const float* __restrict__ Wp3, const float* __restrict__ bp3,
                            float* __restrict__ out, int N) {
    __shared__ float emb[136];
    __shared__ float h1[64];
    __shared__ float h2[32];
    int t = threadIdx.x;

    if (t < 128) emb[t] = scratch[8 + t] / (float)N;
    if (t == 0) {
        float n_comp = scratch[0], n_AND = scratch[1], n_OR = scratch[2];
        float cnt = scratch[3], sl = scratch[4], sm = scratch[5];
        float avg_l = (cnt > 0.0f) ? sl / fmaxf(cnt, 1.0f) : 0.0f;
        float avg_m = (cnt > 0.0f) ? sm / fmaxf(cnt, 1.0f) : 0.0f;
        emb[128] = n_comp; emb[129] = n_AND; emb[130] = n_OR;
        emb[131] = n_AND + n_OR; emb[132] = avg_l; emb[133] = avg_m;
        emb[134] = 100.0f / 500.0f;   // T / T_MAX
    }
    __syncthreads();

    if (t < 135) out[4 + t] = emb[t];

    if (t < 64) {
        float s = bp1[t];
        for (int i = 0; i < 135; ++i) s += emb[i] * Wp1[i * 64 + t];
        h1[t] = fmaxf(s, 0.0f);
    }
    __syncthreads();
    if (t < 32) {
        float s = bp2[t];
        for (int i = 0; i < 64; ++i) s += h1[i] * Wp2[i * 32 + t];
        h2[t] = fmaxf(s, 0.0f);
    }
    __syncthreads();
    if (t == 0) {
        float v[4];
        for (int j = 0; j < 4; ++j) {
            float s = bp3[j];
            for (int i = 0; i < 32; ++i) s += h2[i] * Wp3[i * 4 + j];
            v[j] = (s > 20.0f) ? s : log1pf(expf(s));  // softplus
        }
        const float Tn = 100.0f / 500.0f;
        const float sc = 1.0f - 0.7f * Tn;
        float a_min = 1.0f + v[0] * sc;
        float a_max = 1.0f + (v[0] + v[1] + 1.0f) * sc;
        float b_min = 1.0f + v[2] * sc * 0.3f;
        float b_max = b_min + (v[3] + 0.1f) * sc * 0.3f;
        out[0] = a_min; out[1] = a_max; out[2] = b_min; out[3] = b_max;
    }
}

// ---------------------------------------------------------------------------
extern "C" void kernel_launch(void* const* d_in, const int* in_sizes, int n_in,
                              void* d_out, int out_size, void* d_ws, size_t ws_size,
                              hipStream_t stream) {
    const float* x   = (const float*)d_in[0];
    const int*   ei  = (const int*)d_in[1];
    const float* W1  = (const float*)d_in[2];
    const float* b1  = (const float*)d_in[3];
    const float* W2  = (const float*)d_in[4];
    const float* b2  = (const float*)d_in[5];
    const float* W3  = (const float*)d_in[6];
    const float* b3  = (const float*)d_in[7];
    const float* Wp1 = (const float*)d_in[8];
    const float* bp1 = (const float*)d_in[9];
    const float* Wp2 = (const float*)d_in[10];
    const float* bp2 = (const float*)d_in[11];
    const float* Wp3 = (const float*)d_in[12];
    const float* bp3 = (const float*)d_in[13];

    const int       N = in_sizes[0] / 5;
    const long long E = in_sizes[1] / 2;
    const int* src = ei;
    const int* dst = ei + E;

    // workspace layout (floats)
    float* inv     = (float*)d_ws;
    float* scratch = inv + N;
    float* buf0    = scratch + 1024;
    float* buf1    = buf0 + (size_t)N * 256;

    const int TB = 256;
    auto blocks = [](long long n, int tb) { return (unsigned)((n + tb - 1) / tb); };

    // degrees -> inv = rsqrt(deg+1); gfeat sums; zero scratch
    fill_kernel<<<blocks(N, TB), TB, 0, stream>>>(inv, 0.0f, N);
    fill_kernel<<<blocks(1024, TB), TB, 0, stream>>>(scratch, 0.0f, 1024);
    degree_kernel<<<blocks(E, TB), TB, 0, stream>>>(dst, inv, E);
    rsqrt_kernel<<<blocks(N, TB), TB, 0, stream>>>(inv, N);
    gfeat_reduce<<<256, TB, 0, stream>>>(x, scratch, N);

    const unsigned rowStrips = (unsigned)((N + 31) / 32);

    // ---- layer 1: x[N,5] @ W1[5,256], aggregate, +bias, ReLU ----
    gemm_f32_wmma<<<dim3(rowStrips, 256 / 64), 32, 0, stream>>>(x, W1, buf1, N, 5, 256);
    fill_kernel<<<blocks((long long)N * 256, TB), TB, 0, stream>>>(buf0, 0.0f, (long long)N * 256);
    scatter_kernel<<<blocks(E << 6, TB), TB, 0, stream>>>((const float4*)buf1, src, dst, inv, buf0, E, 6);
    finalize_kernel<<<blocks((long long)N * 256, TB), TB, 0, stream>>>(buf0, buf1, inv, b1, (long long)N * 256, 8, 1);

    // ---- layer 2: h[N,256] @ W2[256,256] ----
    gemm_f32_wmma<<<dim3(rowStrips, 256 / 64), 32, 0, stream>>>(buf0, W2, buf1, N, 256, 256);
    fill_kernel<<<blocks((long long)N * 256, TB), TB, 0, stream>>>(buf0, 0.0f, (long long)N * 256);
    scatter_kernel<<<blocks(E << 6, TB), TB, 0, stream>>>((const float4*)buf1, src, dst, inv, buf0, E, 6);
    finalize_kernel<<<blocks((long long)N * 256, TB), TB, 0, stream>>>(buf0, buf1, inv, b2, (long long)N * 256, 8, 1);

    // ---- layer 3: h[N,256] @ W3[256,128], no ReLU ----
    gemm_f32_wmma<<<dim3(rowStrips, 128 / 64), 32, 0, stream>>>(buf0, W3, buf1, N, 256, 128);
    fill_kernel<<<blocks((long long)N * 128, TB), TB, 0, stream>>>(buf0, 0.0f, (long long)N * 128);
    scatter_kernel<<<blocks(E << 5, TB), TB, 0, stream>>>((const float4*)buf1, src, dst, inv, buf0, E, 5);
    finalize_kernel<<<blocks((long long)N * 128, TB), TB, 0, stream>>>(buf0, buf1, inv, b3, (long long)N * 128, 7, 0);

    // ---- mean pool + head MLP ----
    pool_reduce<<<256, 128, 0, stream>>>(buf0, scratch + 8, N);
    head_kernel<<<1, 160, 0, stream>>>(scratch, Wp1, bp1, Wp2, bp2, Wp3, bp3,
                                       (float*)d_out, N);
}